// KMCL_35553739276539
// MI455X (gfx1250) — compile-verified
//
#include <hip/hip_runtime.h>
#include <hip/hip_bf16.h>

#define EMB_DIM   128
#define N_NODES   200000
#define N_OUT     160000
#define SLOPE     0.01f

typedef float v2f __attribute__((ext_vector_type(2)));
typedef float v8f __attribute__((ext_vector_type(8)));

// ---------------------------------------------------------------------------
// SpMM: one wave32 per edge. Lane L handles dims [4L, 4L+4).
// Gather emb[col] as a coalesced 512B row read, scatter with f32 L2 atomics.
// ---------------------------------------------------------------------------
__global__ void spmm_edge_kernel(const float* __restrict__ vals,
                                 const int*   __restrict__ rows,
                                 const int*   __restrict__ cols,
                                 const float* __restrict__ src,
                                 float*       __restrict__ dst,
                                 int n_edges) {
    int gid  = blockIdx.x * blockDim.x + threadIdx.x;
    int edge = gid >> 5;          // one wave per edge
    int lane = gid & 31;
    if (edge >= n_edges) return;

    float v = vals[edge];
    int   r = rows[edge];
    int   c = cols[edge];

    const float4* s = (const float4*)(src + (size_t)c * EMB_DIM);
    float4 x = s[lane];                                  // dims 4*lane..4*lane+3

    float* d = dst + (size_t)r * EMB_DIM + lane * 4;
    unsafeAtomicAdd(d + 0, v * x.x);
    unsafeAtomicAdd(d + 1, v * x.y);
    unsafeAtomicAdd(d + 2, v * x.z);
    unsafeAtomicAdd(d + 3, v * x.w);
}

// ---------------------------------------------------------------------------
// GEMM: C[m][n] = sum_k side[m][k] * W[n][k]   (i.e. side @ W^T), + bias,
// leaky_relu, fused epilogue. Native f32 WMMA (16x16, K-step 4).
//
// Block = 256 threads = 8 waves. Block b owns rows [16b, 16b+16); wave w owns
// output columns [16w, 16w+16). Per the ISA 16x4 f32 A layout:
//   lane L: M = L%16; VGPR0 holds K = 2*(L/16), VGPR1 holds K = 2*(L/16)+1
// B (4x16) mirrors with N = L%16. C/D: VGPR v -> (M = v + 8*(L/16), N = L%16).
// ---------------------------------------------------------------------------
__global__ void gemm_wmma_kernel(const float* __restrict__ side,
                                 const float* __restrict__ W,
                                 const float* __restrict__ bias,
                                 float*       __restrict__ ego_out,   // layer-0 path
                                 const float* __restrict__ add0,      // embedding
                                 const float* __restrict__ add1,      // ego1
                                 float*       __restrict__ outp,      // final output
                                 int final_mode) {
    const int lane = threadIdx.x & 31;
    const int wave = threadIdx.x >> 5;       // 0..7 -> N tile
    const int rm   = blockIdx.x * 16;        // row-tile base
    const int cn   = wave * 16;              // col-tile base
    const int ml   = lane & 15;
    const int kh   = lane >> 4;              // 0 or 1

    const float* arow = side + (size_t)(rm + ml) * EMB_DIM + 2 * kh;
    const float* brow = W    + (size_t)(cn + ml) * EMB_DIM + 2 * kh;  // B[k][n]=W[n][k]

    v8f acc = {};
    #pragma unroll
    for (int k0 = 0; k0 < EMB_DIM; k0 += 4) {
        v2f a = *(const v2f*)(arow + k0);
        v2f b = *(const v2f*)(brow + k0);
        acc = __builtin_amdgcn_wmma_f32_16x16x4_f32(
            /*neg_a=*/false, a, /*neg_b=*/false, b,
            /*c_mod=*/(short)0, acc, /*reuse_a=*/false, /*reuse_b=*/false);
    }

    const int   cc = cn + ml;                // output column for this lane
    const float bb = bias[cc];

    #pragma unroll
    for (int v = 0; v < 8; ++v) {
        int    r   = rm + v + 8 * kh;
        float  x   = acc[v] + bb;
        x = (x > 0.0f) ? x : SLOPE * x;      // leaky_relu
        size_t idx = (size_t)r * EMB_DIM + cc;
        if (final_mode) {
            // out = (emb + ego1 + ego2) / 3, rows < N_OUT only (grid-limited)
            outp[idx] = (add0[idx] + add1[idx] + x) * (1.0f / 3.0f);
        } else {
            ego_out[idx] = x;
        }
    }
}

// ---------------------------------------------------------------------------
extern "C" void kernel_launch(void* const* d_in, const int* in_sizes, int n_in,
                              void* d_out, int out_size, void* d_ws, size_t ws_size,
                              hipStream_t stream) {
    const float* emb  = (const float*)d_in[0];
    const float* W0   = (const float*)d_in[1];
    const float* b0   = (const float*)d_in[2];
    const float* W1   = (const float*)d_in[3];
    const float* b1   = (const float*)d_in[4];
    const float* vals = (const float*)d_in[5];
    const int*   rows = (const int*)  d_in[6];
    const int*   cols = (const int*)  d_in[7];
    float*       out  = (float*)d_out;

    const int n_edges = in_sizes[5];

    // Workspace: side (N_NODES x 128) | ego1 (N_NODES x 128)   (~205 MB)
    float* side = (float*)d_ws;
    float* ego1 = side + (size_t)N_NODES * EMB_DIM;
    const size_t side_bytes = (size_t)N_NODES * EMB_DIM * sizeof(float);

    dim3 blk(256);
    dim3 spmm_grid((n_edges + 7) / 8);       // 8 edges (waves) per block

    // ----- Layer 0 -----
    hipMemsetAsync(side, 0, side_bytes, stream);
    spmm_edge_kernel<<<spmm_grid, blk, 0, stream>>>(vals, rows, cols, emb, side, n_edges);
    gemm_wmma_kernel<<<N_NODES / 16, blk, 0, stream>>>(
        side, W0, b0, ego1, nullptr, nullptr, nullptr, /*final=*/0);

    // ----- Layer 1 (only first N_OUT rows are needed for the output) -----
    hipMemsetAsync(side, 0, side_bytes, stream);
    spmm_edge_kernel<<<spmm_grid, blk, 0, stream>>>(vals, rows, cols, ego1, side, n_edges);
    gemm_wmma_kernel<<<N_OUT / 16, blk, 0, stream>>>(
        side, W1, b1, nullptr, emb, ego1, out, /*final=*/1);
}